// Model_20675972563286
// MI455X (gfx1250) — compile-verified
//
#include <hip/hip_runtime.h>
#include <hip/hip_bf16.h>
#include <math.h>
#include <float.h>

// ---------------------------------------------------------------------------
// Problem constants (from the reference)
// ---------------------------------------------------------------------------
#define NVV 20000
#define NCC 300
#define NII 10000
#define NNN (NVV + NCC + NII)   // 30300
#define EE  1000000
#define DD  64
#define NTILES (EE / 16)        // 62500 exact (E % 16 == 0)
#define WPB 4                   // waves per block (wave32)
#define TPB (WPB * 32)

typedef __attribute__((ext_vector_type(2))) float v2f;
typedef __attribute__((ext_vector_type(8))) float v8f;

// float atomic max handling negative values (init must be -FLT_MAX)
__device__ __forceinline__ void atomicMaxF(float* a, float v) {
    if (v >= 0.0f) atomicMax((int*)a, __float_as_int(v));
    else           atomicMin((unsigned int*)a, __float_as_uint(v));
}

// ---------------------------------------------------------------------------
// CenterIntersection edge pass.
//  PHASE 0: compute a = MLP(src) per masked edge; atomic-max into mseg[head,d]
//  PHASE 1: recompute a; e = exp(a - m); atomic add sseg += e, nseg += e*src
//  MODE 1: src row = tail[e]           (hist  = all_embs[tail]),   mask = ev
//  MODE 2: src row = vv ? tail : 0     (hist2 = agg2[where(vv,tail,0)]), mask = vv
// One wave handles one 16-edge tile using V_WMMA_F32_16X16X4_F32.
// ---------------------------------------------------------------------------
template <int PHASE, int MODE>
__global__ __launch_bounds__(TPB)
void center_pass(const float* __restrict__ src,
                 const int* __restrict__ head, const int* __restrict__ tail,
                 const float* __restrict__ W1, const float* __restrict__ b1,
                 const float* __restrict__ W2, const float* __restrict__ b2,
                 float* __restrict__ mseg, float* __restrict__ sseg,
                 float* __restrict__ nseg)
{
    __shared__ float lds[WPB][2][16 * DD];   // [wave][0]=src tile, [1]=h tile
    const int wave = threadIdx.x >> 5;
    const int lane = threadIdx.x & 31;
    const int tile = blockIdx.x * WPB + wave;
    if (tile >= NTILES) return;              // uniform per wave; grid is exact
    const int hi   = lane >> 4;              // half-wave selector
    const int lrow = lane & 15;
    const int base = hi << 1;                // K sub-offset {0,2}
    const int e0   = tile << 4;

    // Per-lane head/mask for the 8 D-layout rows this lane owns (M = r + hi*8)
    int  hd[8];
    bool mk[8];
#pragma unroll
    for (int r = 0; r < 8; ++r) {
        int e = e0 + r + hi * 8;
        int h = head[e], t = tail[e];
        hd[r] = h;
        mk[r] = (MODE == 1) ? (h < NVV && t >= NVV)   // ev
                            : (h < NVV && t <  NVV);  // vv
    }

    // A-operand source row: this lane supplies row lrow of the tile
    const int eA = e0 + lrow;
    const int tA = tail[eA];
    const float* srow;
    if (MODE == 1) {
        srow = src + (size_t)tA * DD;
    } else {
        int hA = head[eA];
        int rA = (hA < NVV && tA < NVV) ? tA : 0;
        srow = src + (size_t)rA * DD;
    }

    // Load A-operand (16x4 f32 layout: vgpr0=K(base), vgpr1=K(base+1))
    float a0[16], a1[16];
#pragma unroll
    for (int kc = 0; kc < 16; ++kc) {
        float2 p = *(const float2*)(srow + 4 * kc + base);
        a0[kc] = p.x;
        a1[kc] = p.y;
        if (PHASE == 1) {   // stage src tile for the weighted-sum pass
            lds[wave][0][lrow * DD + 4 * kc + base]     = p.x;
            lds[wave][0][lrow * DD + 4 * kc + base + 1] = p.y;
        }
    }

    // ---- layer 1: h = relu(src @ W1^T + b1) -> LDS --------------------------
#pragma unroll
    for (int nc = 0; nc < 4; ++nc) {
        const int n = nc * 16 + lrow;
        const float bb = b1[n];
        v8f acc = { bb, bb, bb, bb, bb, bb, bb, bb };
        const float* wrow = W1 + (size_t)n * DD;   // B[k][n] = W1[n][k]
#pragma unroll
        for (int kc = 0; kc < 16; ++kc) {
            float2 w = *(const float2*)(wrow + 4 * kc + base);
            v2f A = { a0[kc], a1[kc] };
            v2f B = { w.x, w.y };
            acc = __builtin_amdgcn_wmma_f32_16x16x4_f32(
                false, A, false, B, (short)0, acc, false, false);
        }
#pragma unroll
        for (int r = 0; r < 8; ++r)
            lds[wave][1][(r + hi * 8) * DD + n] = fmaxf(acc[r], 0.0f);
    }

    // wave-synchronous LDS fence: all ds stores visible to this wave's loads
    asm volatile("s_wait_dscnt 0x0" ::: "memory");

    // Reload A-operand for layer 2 from the (transposed-access) h tile
#pragma unroll
    for (int kc = 0; kc < 16; ++kc) {
        a0[kc] = lds[wave][1][lrow * DD + 4 * kc + base];
        a1[kc] = lds[wave][1][lrow * DD + 4 * kc + base + 1];
    }

    // ---- layer 2: a = h @ W2^T + b2, then scatter phase ---------------------
#pragma unroll
    for (int nc = 0; nc < 4; ++nc) {
        const int n = nc * 16 + lrow;
        const float bb = b2[n];
        v8f acc = { bb, bb, bb, bb, bb, bb, bb, bb };
        const float* wrow = W2 + (size_t)n * DD;
#pragma unroll
        for (int kc = 0; kc < 16; ++kc) {
            float2 w = *(const float2*)(wrow + 4 * kc + base);
            v2f A = { a0[kc], a1[kc] };
            v2f B = { w.x, w.y };
            acc = __builtin_amdgcn_wmma_f32_16x16x4_f32(
                false, A, false, B, (short)0, acc, false, false);
        }
#pragma unroll
        for (int r = 0; r < 8; ++r) {
            if (!mk[r]) continue;
            size_t o = (size_t)hd[r] * DD + n;
            if (PHASE == 0) {
                atomicMaxF(&mseg[o], acc[r]);
            } else {
                float ev = __expf(acc[r] - mseg[o]);
                atomicAdd(&sseg[o], ev);
                float sv = lds[wave][0][(r + hi * 8) * DD + n];
                atomicAdd(&nseg[o], ev * sv);
            }
        }
    }
}

// ---------------------------------------------------------------------------
// Elementwise / reduction helpers
// ---------------------------------------------------------------------------
__global__ void k_concat(const float* __restrict__ ve, const float* __restrict__ vo,
                         const float* __restrict__ ce, const float* __restrict__ co,
                         const float* __restrict__ ie, const float* __restrict__ io,
                         float* __restrict__ emb, float* __restrict__ off)
{
    size_t i = (size_t)blockIdx.x * blockDim.x + threadIdx.x;
    if (i >= (size_t)NNN * DD) return;
    size_t row = i / DD, d = i % DD;
    float e, o;
    if (row < NVV)            { e = ve[i]; o = vo[i]; }
    else if (row < NVV + NCC) { size_t j = (row - NVV) * DD + d;       e = ce[j]; o = co[j]; }
    else                      { size_t j = (row - NVV - NCC) * DD + d; e = ie[j]; o = io[j]; }
    emb[i] = e;
    off[i] = fmaxf(o, 0.0f);
}

__global__ void k_time(const float* __restrict__ vt,
                       const float* __restrict__ tW1, const float* __restrict__ tb1,
                       const float* __restrict__ tW2, const float* __restrict__ tb2,
                       float* __restrict__ te)
{
    __shared__ float red[256];
    __shared__ float bc[2];
    const int tid = threadIdx.x;
    const float w1 = tW1[0], bb1 = tb1[0], w2 = tW2[0], bb2 = tb2[0];
    float mx = -FLT_MAX;
    for (int i = tid; i < NVV; i += 256) {
        float t  = 1.0f / vt[i];
        float l1 = fmaxf(t * w1 + bb1, 0.0f);
        mx = fmaxf(mx, l1 * w2 + bb2);
    }
    red[tid] = mx; __syncthreads();
    for (int s = 128; s > 0; s >>= 1) {
        if (tid < s) red[tid] = fmaxf(red[tid], red[tid + s]);
        __syncthreads();
    }
    if (tid == 0) bc[0] = red[0];
    __syncthreads();
    const float mall = bc[0];
    float sm = 0.0f;
    for (int i = tid; i < NVV; i += 256) {
        float t  = 1.0f / vt[i];
        float l1 = fmaxf(t * w1 + bb1, 0.0f);
        sm += __expf(l1 * w2 + bb2 - mall);
    }
    red[tid] = sm; __syncthreads();
    for (int s = 128; s > 0; s >>= 1) {
        if (tid < s) red[tid] += red[tid + s];
        __syncthreads();
    }
    if (tid == 0) bc[1] = red[0];
    __syncthreads();
    const float sall = bc[1];
    for (int i = tid; i < NVV; i += 256) {
        float t  = 1.0f / vt[i];
        float l1 = fmaxf(t * w1 + bb1, 0.0f);
        te[i] = __expf(l1 * w2 + bb2 - mall) / sall;
    }
}

__global__ void k_clear_msn(float* __restrict__ m, float* __restrict__ s,
                            float* __restrict__ n)
{
    size_t i = (size_t)blockIdx.x * blockDim.x + threadIdx.x;
    if (i >= (size_t)NNN * DD) return;
    m[i] = -FLT_MAX;
    s[i] = 0.0f;
    n[i] = 0.0f;
}

__global__ void k_finish_center(const float* __restrict__ s, const float* __restrict__ n,
                                float* __restrict__ agg)
{
    size_t i = (size_t)blockIdx.x * blockDim.x + threadIdx.x;
    if (i >= (size_t)NNN * DD) return;
    float sv = s[i];
    agg[i] = (sv > 0.0f) ? n[i] / sv : 0.0f;
}

__global__ void k_agg2(const float* __restrict__ agg1, const float* __restrict__ te,
                       float* __restrict__ agg2)
{
    size_t i = (size_t)blockIdx.x * blockDim.x + threadIdx.x;
    if (i >= (size_t)NVV * DD) return;
    agg2[i] = agg1[i] * te[i / DD];
}

__global__ void k_l2norm(const float* __restrict__ agg, float* __restrict__ emb)
{
    int row = blockIdx.x * blockDim.x + threadIdx.x;
    if (row >= NNN) return;
    const float* a = agg + (size_t)row * DD;
    float ss = 0.0f;
    for (int d = 0; d < DD; ++d) { float v = a[d]; ss += v * v; }
    float inv = 1.0f / fmaxf(sqrtf(ss), 1e-12f);
    float* e = emb + (size_t)row * DD;
    for (int d = 0; d < DD; ++d) e[d] = a[d] * inv;
}

__global__ void k_clear_off(float* __restrict__ acc)
{
    size_t i = (size_t)blockIdx.x * blockDim.x + threadIdx.x;
    if (i >= (size_t)NNN * DD) return;
    // visit+ccs regions use max (init 0); icd region uses min (init +INF)
    acc[i] = (i < (size_t)(NVV + NCC) * DD) ? 0.0f : __int_as_float(0x7F800000);
}

__global__ void k_off_scatter(const float* __restrict__ off,
                              const int* __restrict__ head, const int* __restrict__ tail,
                              float* __restrict__ acc)
{
    int e = blockIdx.x * blockDim.x + threadIdx.x;
    if (e >= EE) return;
    const int h = head[e], t = tail[e];
    const float* orow = off + (size_t)t * DD;
    float* dst = acc + (size_t)h * DD;
    const bool ismin = (h >= NVV + NCC);
    for (int d = 0; d < DD; ++d) {
        float v = fmaxf(orow[d], 0.0f);      // relu'd => >= 0, int atomics exact
        if (ismin) atomicMin((int*)(dst + d), __float_as_int(v));
        else       atomicMax((int*)(dst + d), __float_as_int(v));
    }
}

__global__ void k_off_finish(const float* __restrict__ acc, float* __restrict__ off)
{
    size_t i = (size_t)blockIdx.x * blockDim.x + threadIdx.x;
    if (i >= (size_t)NNN * DD) return;
    float v = acc[i];
    if (i >= (size_t)(NVV + NCC) * DD && isinf(v)) v = 0.0f;  // empty min segment
    off[i] = fmaxf(v, 0.0f);
}

__global__ void k_out(const float* __restrict__ emb, const float* __restrict__ off,
                      float* __restrict__ out)
{
    size_t i = (size_t)blockIdx.x * blockDim.x + threadIdx.x;
    const size_t half = (size_t)NVV * DD;
    if (i >= 2 * half) return;
    out[i] = (i < half) ? emb[i] : off[i - half];
}

// ---------------------------------------------------------------------------
// Orchestration
// ---------------------------------------------------------------------------
extern "C" void kernel_launch(void* const* d_in, const int* in_sizes, int n_in,
                              void* d_out, int out_size, void* d_ws, size_t ws_size,
                              hipStream_t stream)
{
    const float* visit_emb    = (const float*)d_in[0];
    const float* visit_offset = (const float*)d_in[1];
    const float* ccs_emb      = (const float*)d_in[2];
    const float* ccs_offset   = (const float*)d_in[3];
    const float* icd_emb      = (const float*)d_in[4];
    const float* icd_offset   = (const float*)d_in[5];
    const int*   head         = (const int*)d_in[6];
    const int*   tail         = (const int*)d_in[7];
    const float* visit_time   = (const float*)d_in[8];
    const float* cW1          = (const float*)d_in[9];
    const float* cb1          = (const float*)d_in[10];
    const float* cW2          = (const float*)d_in[11];
    const float* cb2          = (const float*)d_in[12];
    const float* tW1          = (const float*)d_in[13];
    const float* tb1          = (const float*)d_in[14];
    const float* tW2          = (const float*)d_in[15];
    const float* tb2          = (const float*)d_in[16];
    (void)in_sizes; (void)n_in; (void)out_size; (void)ws_size;

    const size_t NN64 = (size_t)NNN * DD;
    float* ws   = (float*)d_ws;
    float* emb  = ws;                        // [NN,64] current embeddings
    float* off  = emb  + NN64;               // [NN,64] current offsets (relu'd)
    float* aggA = off  + NN64;               // [NN,64] center output
    float* agg2 = aggA + NN64;               // [NV,64]
    float* msg  = agg2 + (size_t)NVV * DD;   // [NN,64] segment max
    float* ssg  = msg  + NN64;               // [NN,64] segment sum(e)
    float* nsg  = ssg  + NN64;               // [NN,64] segment sum(e*src)
    float* noff = nsg  + NN64;               // [NN,64] offset accumulators
    float* temb = noff + NN64;               // [NV]    time attention

    const int B = 256;
    const int gNN = (int)((NN64 + B - 1) / B);
    const int gNV = (int)(((size_t)NVV * DD + B - 1) / B);
    const int gE  = (EE + B - 1) / B;
    const int gR  = (NNN + B - 1) / B;
    const int gT  = NTILES / WPB;            // 15625, exact
    const int gO  = (int)((2 * (size_t)NVV * DD + B - 1) / B);

    k_concat<<<gNN, B, 0, stream>>>(visit_emb, visit_offset, ccs_emb, ccs_offset,
                                    icd_emb, icd_offset, emb, off);
    k_time<<<1, 256, 0, stream>>>(visit_time, tW1, tb1, tW2, tb2, temb);

    for (int hop = 0; hop < 2; ++hop) {
        // ---- center 1: entity->visit (mask ev, src = all_embs[tail]) ----
        k_clear_msn<<<gNN, B, 0, stream>>>(msg, ssg, nsg);
        center_pass<0, 1><<<gT, TPB, 0, stream>>>(emb, head, tail, cW1, cb1, cW2, cb2,
                                                  msg, ssg, nsg);
        center_pass<1, 1><<<gT, TPB, 0, stream>>>(emb, head, tail, cW1, cb1, cW2, cb2,
                                                  msg, ssg, nsg);
        k_finish_center<<<gNN, B, 0, stream>>>(ssg, nsg, aggA);
        k_agg2<<<gNV, B, 0, stream>>>(aggA, temb, agg2);

        // ---- center 2: visit->visit (mask vv, src = agg2[where(vv,tail,0)]) ----
        k_clear_msn<<<gNN, B, 0, stream>>>(msg, ssg, nsg);
        center_pass<0, 2><<<gT, TPB, 0, stream>>>(agg2, head, tail, cW1, cb1, cW2, cb2,
                                                  msg, ssg, nsg);
        center_pass<1, 2><<<gT, TPB, 0, stream>>>(agg2, head, tail, cW1, cb1, cW2, cb2,
                                                  msg, ssg, nsg);
        k_finish_center<<<gNN, B, 0, stream>>>(ssg, nsg, aggA);
        k_l2norm<<<gR, B, 0, stream>>>(aggA, emb);

        // ---- box offsets: segmented max/max/min by head region ----
        k_clear_off<<<gNN, B, 0, stream>>>(noff);
        k_off_scatter<<<gE, B, 0, stream>>>(off, head, tail, noff);
        k_off_finish<<<gNN, B, 0, stream>>>(noff, off);
    }

    k_out<<<gO, B, 0, stream>>>(emb, off, (float*)d_out);
}